// GCNEncoder_90486370992276
// MI455X (gfx1250) — compile-verified
//
#include <hip/hip_runtime.h>

typedef float v2f __attribute__((ext_vector_type(2)));
typedef float v8f __attribute__((ext_vector_type(8)));

#define N_NODES 100000
#define N_EDGES 1250000
#define ZBLOCK  256

// ---------------------------------------------------------------------------
// Zero a region (float4 granularity; callers guarantee count % 4 == 0 and
// 16B alignment).
// ---------------------------------------------------------------------------
__global__ __launch_bounds__(ZBLOCK) void gcn_zero_f4(float4* __restrict__ p, int n4) {
    int t = blockIdx.x * ZBLOCK + threadIdx.x;
    if (t < n4) p[t] = make_float4(0.f, 0.f, 0.f, 0.f);
}

// ---------------------------------------------------------------------------
// Degree accumulation: deg[dst[e]] += 1 (f32 atomics into L2-resident buffer)
// ---------------------------------------------------------------------------
__global__ __launch_bounds__(256) void gcn_degree(const int* __restrict__ dst,
                                                  float* __restrict__ deg, int nEdges) {
    int t = blockIdx.x * 256 + threadIdx.x;
    if (t < nEdges) atomicAdd(&deg[dst[t]], 1.0f);
}

// ---------------------------------------------------------------------------
// norm[i] = deg>0 ? rsqrt(max(deg,1)) : 0   (in place: deg buffer -> norm)
// ---------------------------------------------------------------------------
__global__ __launch_bounds__(256) void gcn_norm(float* __restrict__ degnorm, int n) {
    int t = blockIdx.x * 256 + threadIdx.x;
    if (t < n) {
        float d = degnorm[t];
        degnorm[t] = (d > 0.0f) ? rsqrtf(fmaxf(d, 1.0f)) : 0.0f;
    }
}

// ---------------------------------------------------------------------------
// Edge scatter: agg[dst[e], :] += x[src[e], :] * norm[src[e]]
// 16 threads per edge, one float4 feature chunk each (D = 64).
// x and agg are L2-resident (25.6 MB each vs 192 MB L2) -> atomic-bound in L2.
// ---------------------------------------------------------------------------
__global__ __launch_bounds__(256) void gcn_scatter64(const float* __restrict__ x,
                                                     const int* __restrict__ src,
                                                     const int* __restrict__ dst,
                                                     const float* __restrict__ norm,
                                                     float* __restrict__ agg,
                                                     int nEdges) {
    int t = blockIdx.x * 256 + threadIdx.x;
    if (t >= nEdges * 16) return;
    int e = t >> 4;
    int c = t & 15;
    int s = src[e];
    int d = dst[e];
    float ns = norm[s];
    float4 v = ((const float4*)(x + (size_t)s * 64))[c];
    float* a = agg + (size_t)d * 64 + c * 4;
    atomicAdd(a + 0, v.x * ns);
    atomicAdd(a + 1, v.y * ns);
    atomicAdd(a + 2, v.z * ns);
    atomicAdd(a + 3, v.w * ns);
}

// ---------------------------------------------------------------------------
// Fused: out[n,:] = act( (agg[n,:] * norm[n]) @ W + b )
// One 16(M)x16(N) output tile per wave32 using V_WMMA_F32_16X16X4_F32,
// K-loop over DIN in steps of 4 (fully unrolled).
//
// A fragment (16x4 f32, 2 VGPRs): lanes 0-15 hold K=k0,k0+1 for row M=lane;
//                                 lanes 16-31 hold K=k0+2,k0+3 for row M=lane-16.
// B fragment (4x16 f32, 2 VGPRs): mirrored, N = lane&15 across lanes.
// C/D (16x16 f32, 8 VGPRs):       row = r + 8*(lane>=16), col = lane&15.
// Requires nNodes % 16 == 0 (100000 = 6250*16). Wave-uniform guard only, so
// EXEC is all-ones at every WMMA.
// ---------------------------------------------------------------------------
template <int DIN, int DOUT, bool RELU>
__global__ __launch_bounds__(256) void gcn_gemm_wmma(const float* __restrict__ agg,
                                                     const float* __restrict__ norm,
                                                     const float* __restrict__ W,
                                                     const float* __restrict__ bias,
                                                     float* __restrict__ out,
                                                     int nNodes) {
    const int lane = threadIdx.x & 31;
    const int wave = threadIdx.x >> 5;
    const int tilesPerRow = DOUT / 16;
    const int totalTiles  = (nNodes / 16) * tilesPerRow;
    const int tile = blockIdx.x * 8 + wave;     // 8 waves / 256-thread block
    if (tile >= totalTiles) return;             // uniform per wave

    const int nodeBlk = tile / tilesPerRow;
    const int colBlk  = tile % tilesPerRow;
    const int node0   = nodeBlk * 16;
    const int col0    = colBlk * 16;

    const int m  = lane & 15;   // tile row for A / tile col for B,C,D
    const int kh = lane >> 4;   // K half-pair selector for A/B fragments

    const float rowScale = norm[node0 + m];
    const float* arow = agg + (size_t)(node0 + m) * DIN;

    v8f c = {0.f, 0.f, 0.f, 0.f, 0.f, 0.f, 0.f, 0.f};

#pragma unroll
    for (int k0 = 0; k0 < DIN; k0 += 4) {
        const int ka = k0 + 2 * kh;
        // A: two consecutive K values for this lane's row (8B-aligned pair)
        float2 ap = *(const float2*)(arow + ka);
        v2f a;
        a.x = ap.x * rowScale;
        a.y = ap.y * rowScale;
        // B: W is [DIN, DOUT] row-major; rows ka, ka+1, column col0+m
        v2f b;
        b.x = W[(size_t)ka * DOUT + col0 + m];
        b.y = W[(size_t)(ka + 1) * DOUT + col0 + m];
        c = __builtin_amdgcn_wmma_f32_16x16x4_f32(
                /*neg_a=*/false, a, /*neg_b=*/false, b,
                /*c_mod=*/(short)0, c, /*reuse_a=*/false, /*reuse_b=*/false);
    }

    const float bv = bias[col0 + m];
#pragma unroll
    for (int r = 0; r < 8; ++r) {
        const int row = r + 8 * kh;
        float v = c[r] + bv;
        if (RELU) v = fmaxf(v, 0.0f);
        out[(size_t)(node0 + row) * DOUT + col0 + m] = v;
    }
}

// ---------------------------------------------------------------------------
// Host-side orchestration (graph-capture safe: kernels only, single stream).
// Inputs: x[N*64] f32, edge_index[2*E] i32, W1[64*64], b1[64], W2[64*32], b2[32]
// Output: [N*32] f32
// Workspace: norm[N] | agg[N*64] | h[N*64]  (~52 MB)
// ---------------------------------------------------------------------------
extern "C" void kernel_launch(void* const* d_in, const int* in_sizes, int n_in,
                              void* d_out, int out_size, void* d_ws, size_t ws_size,
                              hipStream_t stream) {
    const float* x   = (const float*)d_in[0];
    const int*   ei  = (const int*)d_in[1];
    const float* W1  = (const float*)d_in[2];
    const float* b1  = (const float*)d_in[3];
    const float* W2  = (const float*)d_in[4];
    const float* b2  = (const float*)d_in[5];
    float*       out = (float*)d_out;

    const int N = in_sizes[0] / 64;   // 100000
    const int E = in_sizes[1] / 2;    // 1250000
    const int* src = ei;
    const int* dst = ei + E;

    float* norm = (float*)d_ws;                 // N floats
    float* agg  = norm + N;                     // N*64 floats (16B-aligned: N%4==0)
    float* h    = agg + (size_t)N * 64;         // N*64 floats

    // --- zero norm + agg (contiguous region) ---
    {
        int n4 = (N + N * 64) / 4;
        gcn_zero_f4<<<(n4 + ZBLOCK - 1) / ZBLOCK, ZBLOCK, 0, stream>>>((float4*)norm, n4);
    }

    // --- degree + normalization ---
    gcn_degree<<<(E + 255) / 256, 256, 0, stream>>>(dst, norm, E);
    gcn_norm<<<(N + 255) / 256, 256, 0, stream>>>(norm, N);

    // --- layer 1: scatter(x) -> agg ; wmma gemm (+bias, relu) -> h ---
    {
        int work = E * 16;
        gcn_scatter64<<<(work + 255) / 256, 256, 0, stream>>>(x, src, dst, norm, agg, E);
        int tiles = (N / 16) * (64 / 16);
        gcn_gemm_wmma<64, 64, true><<<(tiles + 7) / 8, 256, 0, stream>>>(agg, norm, W1, b1, h, N);
    }

    // --- re-zero agg for layer 2 ---
    {
        int n4 = (N * 64) / 4;
        gcn_zero_f4<<<(n4 + ZBLOCK - 1) / ZBLOCK, ZBLOCK, 0, stream>>>((float4*)agg, n4);
    }

    // --- layer 2: scatter(h) -> agg ; wmma gemm (+bias) -> out ---
    {
        int work = E * 16;
        gcn_scatter64<<<(work + 255) / 256, 256, 0, stream>>>(h, src, dst, norm, agg, E);
        int tiles = (N / 16) * (32 / 16);
        gcn_gemm_wmma<64, 32, false><<<(tiles + 7) / 8, 256, 0, stream>>>(agg, norm, W2, b2, out, N);
    }
}